// PointEdgeLengthLoss_8117488189443
// MI455X (gfx1250) — compile-verified
//
#include <hip/hip_runtime.h>
#include <cmath>

// MI455X / gfx1250 brute-force KNN (k=16) + edge-length L1 loss.
// Score tiles computed with V_WMMA_F32_16X16X4_F32 (f32, K=4 homogeneous trick).
// All points LDS-resident (128KB of 320KB WGP LDS); selection is register-resident.
// Round 4: make wave/row0/cd0 provably wave-uniform (readfirstlane) so the tile
//          loops compile as scalar-controlled loops instead of divergent loops
//          with exec bookkeeping. Selection stays branchless cndmask chains.

typedef __attribute__((ext_vector_type(2))) float v2f;
typedef __attribute__((ext_vector_type(8))) float v8f;

#define NPTS   8192
#define KNN    16
#define TPB    256
#define WAVES  8
#define TS     18                       // score-tile row stride (floats): conflict-free

#define SPTS_BYTES   (NPTS * 16)        // float4 (x,y,z,|p|^2) per point = 128KB
#define TILE_FLOATS  (32 * TS)          // 32 rows per wave
#define TILE_BYTES   (WAVES * TILE_FLOATS * 4)
#define RED_BYTES    (TPB * 4)
#define SMEM_BYTES   (SPTS_BYTES + TILE_BYTES + RED_BYTES)

__global__ __launch_bounds__(TPB) void knn_edge_loss_kernel(
    const float* __restrict__ pref,     // (B, N, 3) reference points
    const float* __restrict__ ppred,    // (B, N, 3) predicted points
    float* __restrict__ partial)        // one partial sum per block
{
    extern __shared__ char smem[];
    float4* s_pts  = reinterpret_cast<float4*>(smem);                       // [NPTS]
    float*  s_tile = reinterpret_cast<float*>(smem + SPTS_BYTES);           // per-wave tiles
    float*  s_red  = reinterpret_cast<float*>(smem + SPTS_BYTES + TILE_BYTES);

    const int b    = blockIdx.x >> 5;   // 32 blocks per batch
    const int rb   = blockIdx.x & 31;   // row-block (256 rows) within batch
    const int tid  = threadIdx.x;
    const int lane = tid & 31;
    const int half = lane >> 4;         // wave32 half: selects (K0,K1) vs (K2,K3)
    const int l16  = lane & 15;

    // Wave id is uniform across the wave; pin it to an SGPR so row0 / cd0 and
    // the tile-loop trip counts are scalar (s_cmp-controlled loops, no exec
    // bookkeeping per iteration).
    const int wave = __builtin_amdgcn_readfirstlane(tid >> 5);

    const float* Pr = pref  + (size_t)b * NPTS * 3;
    const float* Pp = ppred + (size_t)b * NPTS * 3;

    // Stage the whole batch's ref points into LDS as (x, y, z, |p|^2).
    for (int i = tid; i < NPTS; i += TPB) {
        float x = Pr[3*i+0], y = Pr[3*i+1], z = Pr[3*i+2];
        s_pts[i] = make_float4(x, y, z, x*x + y*y + z*z);
    }
    __syncthreads();

    // This wave owns 32 query rows; this lane owns row m.
    const int row0 = rb * 256 + wave * 32;   // uniform
    const int m    = row0 + lane;

    // A-fragments for V_WMMA_F32_16X16X4_F32 (16x4, M=lane%16):
    // lanes 0-15 hold (K0,K1) = (-2x,-2y); lanes 16-31 hold (K2,K3) = (-2z, 1).
    // With B = (x,y | z,|p|^2), WMMA yields score = |p_n|^2 - 2 q_m.p_n,
    // which is d^2 minus the row-constant |q_m|^2 -> same per-row ordering.
    float4 plo = s_pts[row0 + l16];
    float4 phi = s_pts[row0 + 16 + l16];
    v2f Alo, Ahi;
    if (half == 0) {
        Alo[0] = -2.0f*plo.x; Alo[1] = -2.0f*plo.y;
        Ahi[0] = -2.0f*phi.x; Ahi[1] = -2.0f*phi.y;
    } else {
        Alo[0] = -2.0f*plo.z; Alo[1] = 1.0f;
        Ahi[0] = -2.0f*phi.z; Ahi[1] = 1.0f;
    }

    // Register-resident sorted top-16 (ascending; [15] is the current worst).
    float bestS[KNN]; int bestI[KNN];
    #pragma unroll
    for (int k = 0; k < KNN; ++k) { bestS[k] = 3.4e38f; bestI[k] = 0; }

    // Branchless shift-insert under a single execz-skippable guard.
    // c_t = (s < bestS[t]) is monotone in t; slot t becomes:
    //   c_{t-1} ? bestS[t-1] : (c_t ? s : bestS[t])
    // Pure selects -> v_cndmask chains (heavy dual-issue, all compares
    // independent); one v_cmp per slot shared between adjacent stages by CSE.
    auto insert_guarded = [&](float s, int n, bool self_possible) {
        bool take = (s < bestS[KNN-1]);
        if (self_possible) take = take && (n != m);
        if (take) {
            #pragma unroll
            for (int t = KNN-1; t >= 0; --t) {
                bool cm1 = (t > 0) && (s < bestS[t-1]);
                float ns = cm1 ? bestS[t-1] : s;
                int   ni = cm1 ? bestI[t-1] : n;
                bool ct  = (s < bestS[t]);            // == cm1 of stage t+1 (CSE)
                bestS[t] = ct ? ns : bestS[t];
                bestI[t] = ct ? ni : bestI[t];
            }
        }
    };

    float* tile  = s_tile + wave * TILE_FLOATS;    // private per wave: no barrier needed
    const int rlo = half * 8;                      // C/D layout: rows r (lanes<16) / r+8

    // 8-byte-aligned views for ds_load_b64:
    //  - B-fragment: float2 at &s_pts[n] + half*8B  (float4 entries are 16B-aligned)
    //  - row scan  : lane*TS floats = lane*72B (8B-aligned), tile base 8B-aligned
    const float2* sp2    = reinterpret_cast<const float2*>(s_pts);
    const float2* myrow2 = reinterpret_cast<const float2*>(tile + lane * TS);

    auto do_tile = [&](int c, bool self_possible) {
        // B-fragment: column n = c*16 + lane%16; lanes<16 (x,y), lanes>=16 (z,|p|^2).
        float2 bv = sp2[(c*16 + l16) * 2 + half];  // one ds_load_b64 per lane
        v2f Bf; Bf[0] = bv.x; Bf[1] = bv.y;

        v8f zero = {};
        v8f dlo = __builtin_amdgcn_wmma_f32_16x16x4_f32(
            false, Alo, false, Bf, (short)0, zero, false, false);
        v8f dhi = __builtin_amdgcn_wmma_f32_16x16x4_f32(
            false, Ahi, false, Bf, (short)0, zero, false, false);

        // Scatter 32x16 score tile to LDS (stride 18 -> all 32 lanes hit distinct
        // banks; compiler pairs these into ds_store_2addr_b32).
        #pragma unroll
        for (int r = 0; r < 8; ++r) tile[(rlo + r)      * TS + l16] = dlo[r];
        #pragma unroll
        for (int r = 0; r < 8; ++r) tile[(16 + rlo + r) * TS + l16] = dhi[r];
        // DS ops are in-order within a wave; compiler inserts s_wait_dscnt before use.

        const int nb = c * 16;
        #pragma unroll
        for (int j2 = 0; j2 < 8; ++j2) {           // 8 x ds_load_b64 row scan
            float2 sv = myrow2[j2];
            insert_guarded(sv.x, nb + 2*j2,     self_possible);
            insert_guarded(sv.y, nb + 2*j2 + 1, self_possible);
        }
    };

    // n == m only possible in the 2 tiles covering this wave's own 32 rows;
    // all other 510 tiles skip the self-exclusion compare entirely.
    const int cd0 = row0 >> 4;                     // first diagonal tile (uniform)
    for (int c = 0;       c < cd0;      ++c) do_tile(c, false);
    do_tile(cd0,     true);
    do_tile(cd0 + 1, true);
    for (int c = cd0 + 2; c < NPTS/16; ++c) do_tile(c, false);

    // Edge-length L1 for this lane's row: ref neighbors from LDS, pred from global (L2-hot).
    __builtin_prefetch(&Pp[3*m], 0, 0);
    float4 qr = s_pts[m];
    float qx = Pp[3*m+0], qy = Pp[3*m+1], qz = Pp[3*m+2];
    float acc = 0.0f;
    #pragma unroll
    for (int k = 0; k < KNN; ++k) {
        int i = bestI[k];
        float4 nr = s_pts[i];
        float dx = nr.x - qr.x, dy = nr.y - qr.y, dz = nr.z - qr.z;
        float dref = sqrtf(dx*dx + dy*dy + dz*dz);
        float ex = Pp[3*i+0] - qx, ey = Pp[3*i+1] - qy, ez = Pp[3*i+2] - qz;
        float dprd = sqrtf(ex*ex + ey*ey + ez*ez);
        acc += fabsf(dref - dprd);
    }

    s_red[tid] = acc;
    __syncthreads();
    if (tid == 0) {
        float t = 0.0f;
        for (int i = 0; i < TPB; ++i) t += s_red[i];
        partial[blockIdx.x] = t;
    }
}

// Deterministic final reduction (no float atomics -> bit-stable across graph replays).
__global__ void finalize_kernel(const float* __restrict__ partial,
                                float* __restrict__ out, int nblocks, float scale)
{
    if (threadIdx.x == 0 && blockIdx.x == 0) {
        float t = 0.0f;
        for (int i = 0; i < nblocks; ++i) t += partial[i];
        out[0] = t * scale;
    }
}

extern "C" void kernel_launch(void* const* d_in, const int* in_sizes, int n_in,
                              void* d_out, int out_size, void* d_ws, size_t ws_size,
                              hipStream_t stream)
{
    const float* pref  = (const float*)d_in[0];   // points_ref (B,N,3) f32
    const float* ppred = (const float*)d_in[1];   // points     (B,N,3) f32
    float* out     = (float*)d_out;
    float* partial = (float*)d_ws;                // nblocks floats of scratch

    const int B       = in_sizes[0] / (NPTS * 3); // = 4
    const int nblocks = B * 32;

    // Allow >64KB dynamic LDS (gfx1250 WGP has 320KB). Idempotent, not a stream op.
    (void)hipFuncSetAttribute(reinterpret_cast<const void*>(knn_edge_loss_kernel),
                              hipFuncAttributeMaxDynamicSharedMemorySize, SMEM_BYTES);

    knn_edge_loss_kernel<<<nblocks, TPB, SMEM_BYTES, stream>>>(pref, ppred, partial);

    const float scale = 1.0f / (float)(B * NPTS * KNN);
    finalize_kernel<<<1, 32, 0, stream>>>(partial, out, nblocks, scale);
}